// CategoricalLayer_37512244363977
// MI455X (gfx1250) — compile-verified
//
#include <hip/hip_runtime.h>
#include <hip/hip_bf16.h>

#ifndef __has_builtin
#define __has_builtin(x) 0
#endif

#if __has_builtin(__builtin_amdgcn_global_load_async_to_lds_b128) && \
    __has_builtin(__builtin_amdgcn_s_wait_asynccnt)
#define USE_ASYNC_LDS 1
#else
#define USE_ASYNC_LDS 0
#endif

typedef float v4f __attribute__((ext_vector_type(4)));
typedef int   v4i __attribute__((ext_vector_type(4)));

namespace {
constexpr int kV             = 64;
constexpr int kNPerVar       = 512;
constexpr int kC             = 256;              // categories per node
constexpr int kB             = 1024;             // batch
constexpr int kNodes         = kV * kNPerVar;    // 32768
constexpr int kNodesPerBlock = 16;               // 16 | 512 -> one variable per block
constexpr int kThreads       = 256;              // 8 wave32
constexpr int kWaves         = kThreads / 32;    // 8
constexpr int kNodesPerWave  = kNodesPerBlock / kWaves;  // 2
constexpr float kEpsAdd      = 1e-8f;
constexpr float kEpsClamp    = 1e-10f;
}

#if USE_ASYNC_LDS
// Async copy 16 bytes global -> LDS (CDNA5 GLOBAL_LOAD_ASYNC_TO_LDS_B128,
// tracked by ASYNCcnt; data never touches VGPRs).
__device__ __forceinline__ void async_ld_b128(const void* gp, void* lp) {
  __builtin_amdgcn_global_load_async_to_lds_b128(
      (__attribute__((address_space(1))) v4i*)(void*)gp,
      (__attribute__((address_space(3))) v4i*)lp,
      /*offset=*/0, /*cpol=*/0);
}
#endif

// log(max(p + 1e-8, 1e-10)); argument is >= 1e-10 so no denormal fixups needed:
// raw v_log_f32 (log2) * ln2 is exact enough and only 4 VALU ops per element.
__device__ __forceinline__ float cat_log(float p) {
  const float x = fmaxf(p + kEpsAdd, kEpsClamp);
#if __has_builtin(__builtin_amdgcn_logf)
  return __builtin_amdgcn_logf(x) * 0.69314718055994530942f;
#else
  return __logf(x);
#endif
}

__global__ __launch_bounds__(kThreads)
void cat_log_kernel(const int* __restrict__ data,     // [V, B]
                    const float* __restrict__ params, // [kNodes * kC]
                    const int* __restrict__ vids,     // [kNodes]
                    const int* __restrict__ psids,    // [kNodes]
                    float* __restrict__ out) {        // [kNodes, B]
  __shared__ int   s_data[kB];                  //  4 KB: category ids, this variable
  __shared__ float s_par[kNodesPerBlock * kC];  // 16 KB: 16 param blocks

  const int tid = threadIdx.x;
  const int n0  = blockIdx.x * kNodesPerBlock;
  const int var = vids[n0];  // all 16 nodes of this block share one variable
  const int* __restrict__ drow = data + var * kB;

#if USE_ASYNC_LDS
  // Data row: 1024 ints = 256 x b128, one async per thread. Branch-free.
  async_ld_b128(drow + tid * 4, &s_data[tid * 4]);

  // Param blocks: each wave stages 2 whole nodes (64 b128 chunks per node,
  // 32 lanes x 2 chunks). psid is wave-uniform -> readfirstlane puts the
  // global base in SGPRs (GVS addressing for the async load).
  const int wave = tid >> 5;   // 0..7, uniform within a wave32
  const int lane = tid & 31;
  #pragma unroll
  for (int i = 0; i < kNodesPerWave; ++i) {
    const int ln = wave * kNodesPerWave + i;
    const int psid = __builtin_amdgcn_readfirstlane(psids[n0 + ln]);
    const float* __restrict__ pb = params + psid;
    float* lb = &s_par[ln * kC];
    async_ld_b128(pb + lane * 4,        lb + lane * 4);
    async_ld_b128(pb + (lane + 32) * 4, lb + (lane + 32) * 4);
  }
  __builtin_amdgcn_s_wait_asynccnt(0);  // this wave's async copies done in LDS
  __syncthreads();                      // visible workgroup-wide
#else
  // Fallback: VGPR-bounced staging (global_load_b128 + ds_store_b128).
  ((v4i*)s_data)[tid] = ((const v4i*)drow)[tid];
  #pragma unroll
  for (int k = tid; k < kNodesPerBlock * (kC / 4); k += kThreads) {
    const int ln = k >> 6;
    const int j  = k & 63;
    *(v4f*)&s_par[ln * kC + j * 4] =
        *(const v4f*)(params + psids[n0 + ln] + j * 4);
  }
  __syncthreads();
#endif

  // Each thread owns 4 consecutive batch elements; its 4 category ids are
  // reused across all 16 nodes. LDS gather + v_log_f32, NT b128 stores.
  const int b0 = tid * 4;
  const v4i c = *(const v4i*)&s_data[b0];

  #pragma unroll
  for (int ln = 0; ln < kNodesPerBlock; ++ln) {
    const float* __restrict__ pb = &s_par[ln * kC];
    v4f r;
    r.x = cat_log(pb[c.x]);
    r.y = cat_log(pb[c.y]);
    r.z = cat_log(pb[c.z]);
    r.w = cat_log(pb[c.w]);
    v4f* dst = (v4f*)(out + (size_t)(n0 + ln) * kB + b0);
    __builtin_nontemporal_store(r, dst);  // 134 MB stream: keep it out of L2
  }
}

extern "C" void kernel_launch(void* const* d_in, const int* in_sizes, int n_in,
                              void* d_out, int out_size, void* d_ws, size_t ws_size,
                              hipStream_t stream) {
  (void)in_sizes; (void)n_in; (void)out_size; (void)d_ws; (void)ws_size;
  const int*   data   = (const int*)d_in[0];
  const float* params = (const float*)d_in[1];
  const int*   vids   = (const int*)d_in[2];
  const int*   psids  = (const int*)d_in[3];
  float*       out    = (float*)d_out;

  dim3 grid(kNodes / kNodesPerBlock);  // 2048 blocks
  dim3 block(kThreads);                // 256 threads = 8 wave32
  cat_log_kernel<<<grid, block, 0, stream>>>(data, params, vids, psids, out);
}